// GCN_MLP_23957327577458
// MI455X (gfx1250) — compile-verified
//
#include <hip/hip_runtime.h>

typedef __attribute__((ext_vector_type(2))) float v2f;
typedef __attribute__((ext_vector_type(8))) float v8f;

// ---------------------------------------------------------------- utilities
__global__ void k_zero(float* __restrict__ p, long long n) {
  long long i = (long long)blockIdx.x * blockDim.x + threadIdx.x;
  long long stride = (long long)gridDim.x * blockDim.x;
  for (; i < n; i += stride) p[i] = 0.0f;
}

// deg[col[e]] += ew[e]  (self-loop +1 folded into k_dinv)
__global__ void k_deg(const long long* __restrict__ ei, const float* __restrict__ ew,
                      float* __restrict__ deg, long long E) {
  long long e = (long long)blockIdx.x * blockDim.x + threadIdx.x;
  if (e >= E) return;
  long long c = ei[E + e];
  atomicAdd(&deg[c], ew[e]);
}

// dinv[i] = rsqrt(deg[i] + 1)   (in place)
__global__ void k_dinv(float* __restrict__ d, int n) {
  int i = blockIdx.x * blockDim.x + threadIdx.x;
  if (i >= n) return;
  float v = d[i] + 1.0f;
  d[i] = v > 0.0f ? rsqrtf(v) : 0.0f;
}

// ---------------------------------------------------------------- WMMA GEMM
// out[nRows,32] = feat[nRows,K] @ W[K,32], fp32 WMMA 16x16x4.
// One 16-row tile per wave, two 16x16 WMMAs per K-step (N 0..15 / 16..31).
template <int K>
__global__ __launch_bounds__(128) void k_gemm(const float* __restrict__ feat,
                                              const float* __restrict__ W,
                                              float* __restrict__ out, int nRows) {
  __shared__ float sW[K * 32];
  for (int i = threadIdx.x; i < K * 32; i += 128) sW[i] = W[i];
  __syncthreads();

  const int lane = threadIdx.x & 31;
  const int wave = threadIdx.x >> 5;
  const int tile = blockIdx.x * 4 + wave;          // wave-uniform
  if (tile * 16 >= nRows) return;                  // EXEC stays all-ones per wave

  const int mL   = lane & 15;
  const int kOff = (lane >> 4) * 2;                // lanes 0-15 -> K pair 0,1 ; 16-31 -> 2,3
  const float* aRow = feat + (size_t)(tile * 16 + mL) * K;

  v8f acc0 = {};
  v8f acc1 = {};
#pragma unroll 4
  for (int k = 0; k < K; k += 4) {
    v2f a;
    a[0] = aRow[k + kOff];
    a[1] = aRow[k + kOff + 1];
    v2f b0, b1;
    b0[0] = sW[(k + kOff) * 32 + mL];
    b0[1] = sW[(k + kOff + 1) * 32 + mL];
    b1[0] = sW[(k + kOff) * 32 + 16 + mL];
    b1[1] = sW[(k + kOff + 1) * 32 + 16 + mL];
    acc0 = __builtin_amdgcn_wmma_f32_16x16x4_f32(false, a, false, b0, (short)0, acc0, false, false);
    acc1 = __builtin_amdgcn_wmma_f32_16x16x4_f32(false, a, false, b1, (short)0, acc1, false, false);
  }

  // C/D layout: VGPR r -> row (r + 8*(lane>>4)), col lane&15
  const int rBase = tile * 16 + (lane >> 4) * 8;
#pragma unroll
  for (int r = 0; r < 8; ++r) {
    out[(size_t)(rBase + r) * 32 + mL]      = acc0[r];
    out[(size_t)(rBase + r) * 32 + 16 + mL] = acc1[r];
  }
}

// ------------------------------------------------------- edge scatter-gather
// 8 lanes per edge, float4 per lane; h (12.8 MB) is L2-resident on MI455X.
__global__ __launch_bounds__(256) void k_edge(const long long* __restrict__ ei,
                                              const float* __restrict__ ew,
                                              const float* __restrict__ dinv,
                                              const float* __restrict__ hlin,
                                              float* __restrict__ hagg, long long E) {
  long long t = (long long)blockIdx.x * blockDim.x + threadIdx.x;
  long long e = t >> 3;
  if (e >= E) return;
  int f = (int)(t & 7) * 4;
  long long r = ei[e];
  long long c = ei[E + e];
  float coeff = dinv[r] * ew[e] * dinv[c];
  float4 v = *(const float4*)(hlin + r * 32 + f);
  float* dst = hagg + c * 32 + f;
  atomicAdd(dst + 0, coeff * v.x);
  atomicAdd(dst + 1, coeff * v.y);
  atomicAdd(dst + 2, coeff * v.z);
  atomicAdd(dst + 3, coeff * v.w);
}

// out = hagg + dinv[i]^2 * hlin + bias  (self-loop fused), optional ReLU, in place
__global__ void k_finish(float* __restrict__ hagg, const float* __restrict__ hlin,
                         const float* __restrict__ dinv, const float* __restrict__ bias,
                         int n, int do_relu) {
  long long t = (long long)blockIdx.x * blockDim.x + threadIdx.x;
  if (t >= (long long)n * 32) return;
  int i = (int)(t >> 5), f = (int)(t & 31);
  float di = dinv[i];
  float v = hagg[t] + di * di * hlin[t] + bias[f];
  hagg[t] = do_relu ? fmaxf(v, 0.0f) : v;
}

// ---------------------------------------------------------------- pooling
__global__ void k_pool(const float* __restrict__ h, const long long* __restrict__ batch,
                       float* __restrict__ sums, float* __restrict__ counts, int n) {
  long long t = (long long)blockIdx.x * blockDim.x + threadIdx.x;
  if (t >= (long long)n * 32) return;
  int i = (int)(t >> 5), f = (int)(t & 31);
  int b = (int)batch[i];
  atomicAdd(&sums[b * 32 + f], h[t]);
  if (f == 0) atomicAdd(&counts[b], 1.0f);
}

// ---------------------------------------------------------------- fused MLP
__global__ __launch_bounds__(256) void k_mlp(const float* __restrict__ sums,
                                             const float* __restrict__ counts,
                                             const float* __restrict__ Wm0, const float* __restrict__ bm0,
                                             const float* __restrict__ Wm1, const float* __restrict__ bm1,
                                             const float* __restrict__ Wout, const float* __restrict__ bout,
                                             float* __restrict__ out) {
  __shared__ float gA[64 * 64];
  __shared__ float gB[64 * 64];
  int tid = threadIdx.x;
  // mean pool -> gA[64x32]
  for (int idx = tid; idx < 64 * 32; idx += 256) {
    int r = idx >> 5;
    gA[idx] = sums[idx] / fmaxf(counts[r], 1.0f);
  }
  __syncthreads();
  // gB = relu(gA[64x32] @ Wm0[32x64] + bm0)
  for (int idx = tid; idx < 64 * 64; idx += 256) {
    int r = idx >> 6, c = idx & 63;
    float acc = bm0[c];
    for (int k = 0; k < 32; ++k) acc += gA[r * 32 + k] * Wm0[k * 64 + c];
    gB[idx] = fmaxf(acc, 0.0f);
  }
  __syncthreads();
  // gA = relu(gB @ Wm1[64x64] + bm1)
  for (int idx = tid; idx < 64 * 64; idx += 256) {
    int r = idx >> 6, c = idx & 63;
    float acc = bm1[c];
    for (int k = 0; k < 64; ++k) acc += gB[r * 64 + k] * Wm1[k * 64 + c];
    gA[idx] = fmaxf(acc, 0.0f);
  }
  __syncthreads();
  // out = gA @ Wout[64x10] + bout
  for (int idx = tid; idx < 64 * 10; idx += 256) {
    int r = idx / 10, c = idx % 10;
    float acc = bout[c];
    for (int k = 0; k < 64; ++k) acc += gA[r * 64 + k] * Wout[k * 10 + c];
    out[idx] = acc;
  }
}

// ---------------------------------------------------------------- driver
extern "C" void kernel_launch(void* const* d_in, const int* in_sizes, int n_in,
                              void* d_out, int out_size, void* d_ws, size_t ws_size,
                              hipStream_t stream) {
  const float*     x     = (const float*)d_in[0];
  const long long* ei    = (const long long*)d_in[1];
  const float*     ew    = (const float*)d_in[2];
  const long long* batch = (const long long*)d_in[3];
  const float *W1 = (const float*)d_in[4],  *b1 = (const float*)d_in[5];
  const float *W2 = (const float*)d_in[6],  *b2 = (const float*)d_in[7];
  const float *W3 = (const float*)d_in[8],  *b3 = (const float*)d_in[9];
  const float *Wm0 = (const float*)d_in[10], *bm0 = (const float*)d_in[11];
  const float *Wm1 = (const float*)d_in[12], *bm1 = (const float*)d_in[13];
  const float *Wo  = (const float*)d_in[14], *bo  = (const float*)d_in[15];
  float* out = (float*)d_out;

  const long long E = in_sizes[2];   // edge_weight count
  const int       N = in_sizes[3];   // batch count = #nodes

  // workspace layout (floats)
  float* ws = (float*)d_ws;
  size_t nh = (size_t)N * 32;
  float* A    = ws;            // linear-transform output (per layer)
  float* B    = A + nh;        // agg / feature ping
  float* Cb   = B + nh;        // agg / feature pong
  float* dinv = Cb + nh;       // N
  float* sums = dinv + N;      // 64*32
  float* counts = sums + 64 * 32; // 64

  const int T = 256;
  const int tiles   = (N + 15) / 16;
  const int gblocks = (tiles + 3) / 4;
  const int eblocks = (int)((E * 8 + T - 1) / T);
  const int fblocks = (int)(((long long)N * 32 + T - 1) / T);

  // symmetric normalization coefficients
  k_zero<<<1024, T, 0, stream>>>(dinv, N);
  k_deg<<<(int)((E + T - 1) / T), T, 0, stream>>>(ei, ew, dinv, E);
  k_dinv<<<(N + T - 1) / T, T, 0, stream>>>(dinv, N);

  // layer 1: x (K=128) -> lin A, agg B, finish in place (ReLU)
  k_gemm<128><<<gblocks, 128, 0, stream>>>(x, W1, A, N);
  k_zero<<<2048, T, 0, stream>>>(B, (long long)nh);
  k_edge<<<eblocks, T, 0, stream>>>(ei, ew, dinv, A, B, E);
  k_finish<<<fblocks, T, 0, stream>>>(B, A, dinv, b1, N, 1);

  // layer 2: B (K=32) -> lin A, agg Cb (ReLU)
  k_gemm<32><<<gblocks, 128, 0, stream>>>(B, W2, A, N);
  k_zero<<<2048, T, 0, stream>>>(Cb, (long long)nh);
  k_edge<<<eblocks, T, 0, stream>>>(ei, ew, dinv, A, Cb, E);
  k_finish<<<fblocks, T, 0, stream>>>(Cb, A, dinv, b2, N, 1);

  // layer 3: Cb (K=32) -> lin A, agg B (no ReLU)
  k_gemm<32><<<gblocks, 128, 0, stream>>>(Cb, W3, A, N);
  k_zero<<<2048, T, 0, stream>>>(B, (long long)nh);
  k_edge<<<eblocks, T, 0, stream>>>(ei, ew, dinv, A, B, E);
  k_finish<<<fblocks, T, 0, stream>>>(B, A, dinv, b3, N, 0);

  // global mean pool + MLP head
  k_zero<<<1, T, 0, stream>>>(sums, 64 * 32 + 64);
  k_pool<<<fblocks, T, 0, stream>>>(B, batch, sums, counts, N);
  k_mlp<<<1, 256, 0, stream>>>(sums, counts, Wm0, bm0, Wm1, bm1, Wo, bo, out);
}